// KoopmanOperators_3015067041822
// MI455X (gfx1250) — compile-verified
//
#include <hip/hip_runtime.h>
#include <math.h>

typedef float v2f __attribute__((ext_vector_type(2)));
typedef float v8f __attribute__((ext_vector_type(8)));

namespace {
constexpr int Bb = 16, Nn = 24, Tt = 16, SD = 16;
constexpr int G = 64, R = 64, U = 16, C = 4;
constexpr int F = G + R + U;                                  // 144
constexpr size_t OUT_ELEMS = (size_t)Bb * Nn * Tt * C * F;    // 3,538,944
constexpr size_t SEL_ELEMS = (size_t)Bb * Nn * Tt * Nn;       // 147,456
}

__device__ __forceinline__ float relu_f(float x) { return x > 0.f ? x : 0.f; }

// Wave-level GEMM: 2 row-tiles x 2 col-tiles (32x32 output patch) of
// D = A(32xK, LDS row-major stride lda) * B(KxN, global row-major stride ldb).
// A fragments are reused across both col-tiles; B fragments across both row-tiles:
// per K-chunk: 2 LDS loads + 4 global loads feed 4 WMMAs.
// f32 WMMA fragment layout (ISA 7.12.2): A lanes0-15 K={k,k+1}, lanes16-31 K={k+2,k+3};
// B mirrored with lanes indexing N; C/D row = r + 8*(lane>=16), col = lane&15.
template <int K>
__device__ __forceinline__ void wave_gemm22(const float* lA, int lda,
                                            const float* __restrict__ Bg, int ldb,
                                            int col0, v8f& c00, v8f& c01,
                                            v8f& c10, v8f& c11) {
  const int lane = threadIdx.x & 31;
  const int m = lane & 15;
  const int kh = (lane >> 4) << 1;  // 0 or 2
  const float* a0p = lA + m * lda;
  const float* a1p = lA + (m + 16) * lda;
  const float* bp0 = Bg + col0 + m;  // lane indexes column n == m
  const float* bp1 = bp0 + 16;
#pragma unroll 4
  for (int k0 = 0; k0 < K; k0 += 4) {
    const int k = k0 + kh;
    v2f a0 = *(const v2f*)(a0p + k);
    v2f a1 = *(const v2f*)(a1p + k);
    v2f b0, b1;
    b0.x = bp0[(size_t)k * ldb];
    b0.y = bp0[(size_t)(k + 1) * ldb];
    b1.x = bp1[(size_t)k * ldb];
    b1.y = bp1[(size_t)(k + 1) * ldb];
    c00 = __builtin_amdgcn_wmma_f32_16x16x4_f32(false, a0, false, b0, (short)0,
                                                c00, false, false);
    c10 = __builtin_amdgcn_wmma_f32_16x16x4_f32(false, a1, false, b0, (short)0,
                                                c10, false, false);
    c01 = __builtin_amdgcn_wmma_f32_16x16x4_f32(false, a0, false, b1, (short)0,
                                                c01, false, false);
    c11 = __builtin_amdgcn_wmma_f32_16x16x4_f32(false, a1, false, b1, (short)0,
                                                c11, false, false);
  }
}

// bias + ReLU epilogue of one 32x16 col-tile into LDS (stride ldd)
__device__ __forceinline__ void epi_relu_lds(float* dst, int ldd, int col0,
                                             const float* __restrict__ bias,
                                             const v8f& lo, const v8f& hi) {
  const int lane = threadIdx.x & 31;
  const int n = lane & 15;
  const int rb = (lane >> 4) * 8;
  const float bb = bias[col0 + n];
#pragma unroll
  for (int r = 0; r < 8; ++r) {
    dst[(rb + r) * ldd + col0 + n] = relu_f(lo[r] + bb);
    dst[(16 + rb + r) * ldd + col0 + n] = relu_f(hi[r] + bb);
  }
}

// ---------------- node embedding + U path: one block per bt ----------------
__global__ __launch_bounds__(256) void koop_node_kernel(
    const float* __restrict__ states,
    const float* __restrict__ e1w, const float* __restrict__ e1b,
    const float* __restrict__ e2w, const float* __restrict__ e2b,
    const float* __restrict__ e3w, const float* __restrict__ e3b,
    const float* __restrict__ e4w, const float* __restrict__ e4b,
    const float* __restrict__ us1w, const float* __restrict__ us1b,
    const float* __restrict__ us2w, const float* __restrict__ us2b,
    const float* __restrict__ um1w, const float* __restrict__ um1b,
    const float* __restrict__ um2w, const float* __restrict__ um2b,
    const float* __restrict__ um3w, const float* __restrict__ um3b,
    float* __restrict__ out, float* __restrict__ u_out) {
  __shared__ __align__(16) float xs[32 * 16];
  __shared__ __align__(16) float hA[32 * 512];
  __shared__ __align__(16) float hB[32 * 512];
  __shared__ __align__(16) float red[256];

  const int tid = threadIdx.x;
  const int wave = tid >> 5;
  const int lane = tid & 31;
  const int n16 = lane & 15;
  const int rb = (lane >> 4) * 8;
  const int bt = blockIdx.x;
  const int b = bt >> 4, t = bt & 15;

  for (int idx = tid; idx < 32 * 16; idx += 256) {
    int n = idx >> 4, sd = idx & 15;
    xs[idx] = (n < Nn) ? states[((size_t)(b * Nn + n) * Tt + t) * SD + sd] : 0.f;
  }
  __syncthreads();
  for (int p = wave; p < 16; p += 8) {               // emb1: 16 -> 512
    v8f c00 = {}, c01 = {}, c10 = {}, c11 = {};
    wave_gemm22<16>(xs, 16, e1w, 512, p * 32, c00, c01, c10, c11);
    epi_relu_lds(hA, 512, p * 32, e1b, c00, c10);
    epi_relu_lds(hA, 512, p * 32 + 16, e1b, c01, c11);
  }
  __syncthreads();
  for (int p = wave; p < 16; p += 8) {               // emb2: 512 -> 512
    v8f c00 = {}, c01 = {}, c10 = {}, c11 = {};
    wave_gemm22<512>(hA, 512, e2w, 512, p * 32, c00, c01, c10, c11);
    epi_relu_lds(hB, 512, p * 32, e2b, c00, c10);
    epi_relu_lds(hB, 512, p * 32 + 16, e2b, c01, c11);
  }
  __syncthreads();
  for (int p = wave; p < 16; p += 8) {               // emb3: 512 -> 512
    v8f c00 = {}, c01 = {}, c10 = {}, c11 = {};
    wave_gemm22<512>(hB, 512, e3w, 512, p * 32, c00, c01, c10, c11);
    epi_relu_lds(hA, 512, p * 32, e3b, c00, c10);
    epi_relu_lds(hA, 512, p * 32 + 16, e3b, c01, c11);
  }
  __syncthreads();
  for (int p = wave; p < 8; p += 8) {                // emb4 cols 1..256 -> g
    v8f c00 = {}, c01 = {}, c10 = {}, c11 = {};
    wave_gemm22<512>(hA, 512, e4w + 1, G * C + 1, p * 32, c00, c01, c10, c11);
    auto write_g = [&](const v8f& lo, const v8f& hi, int col) {
      const int c = col >> 6, gi = col & 63;
      const float bb = e4b[1 + col];
#pragma unroll
      for (int r = 0; r < 8; ++r) {
        int row = rb + r;                                           // < 16
        out[(((size_t)(b * Nn + row) * Tt + t) * C + c) * F + gi] = lo[r] + bb;
        row += 16;
        if (row < Nn)
          out[(((size_t)(b * Nn + row) * Tt + t) * C + c) * F + gi] = hi[r] + bb;
      }
    };
    write_g(c00, c10, p * 32 + n16);
    write_g(c01, c11, p * 32 + 16 + n16);
  }
  for (int p = wave; p < 8; p += 8) {                // usel1: 16 -> 256
    v8f c00 = {}, c01 = {}, c10 = {}, c11 = {};
    wave_gemm22<16>(xs, 16, us1w, 256, p * 32, c00, c01, c10, c11);
    epi_relu_lds(hB, 256, p * 32, us1b, c00, c10);
    epi_relu_lds(hB, 256, p * 32 + 16, us1b, c01, c11);
  }
  __syncthreads();
  {                                                   // usel2 dot -> sigmoid gate
    int row = tid >> 3, seg = tid & 7;
    const float* hr = hB + row * 256 + seg * 32;
    const float* wr = us2w + seg * 32;
    float s = 0.f;
#pragma unroll
    for (int q = 0; q < 32; ++q) s += hr[q] * wr[q];
    red[tid] = s;
  }
  __syncthreads();
  if (tid < 32) {
    float tot = us2b[0];
#pragma unroll
    for (int q = 0; q < 8; ++q) tot += red[tid * 8 + q];
    if (tid < Nn) u_out[(size_t)(b * Nn + tid) * Tt + t] = 1.f / (1.f + expf(-tot));
  }
  __syncthreads();
  for (int p = wave; p < 16; p += 8) {               // um1: 16 -> 512
    v8f c00 = {}, c01 = {}, c10 = {}, c11 = {};
    wave_gemm22<16>(xs, 16, um1w, 512, p * 32, c00, c01, c10, c11);
    epi_relu_lds(hA, 512, p * 32, um1b, c00, c10);
    epi_relu_lds(hA, 512, p * 32 + 16, um1b, c01, c11);
  }
  __syncthreads();
  for (int p = wave; p < 16; p += 8) {               // um2: 512 -> 512
    v8f c00 = {}, c01 = {}, c10 = {}, c11 = {};
    wave_gemm22<512>(hA, 512, um2w, 512, p * 32, c00, c01, c10, c11);
    epi_relu_lds(hB, 512, p * 32, um2b, c00, c10);
    epi_relu_lds(hB, 512, p * 32 + 16, um2b, c01, c11);
  }
  __syncthreads();
  for (int p = wave; p < 2; p += 8) {                // um3: 512 -> 64, tanh -> g_u
    v8f c00 = {}, c01 = {}, c10 = {}, c11 = {};
    wave_gemm22<512>(hB, 512, um3w, U * C, p * 32, c00, c01, c10, c11);
    auto write_u = [&](const v8f& lo, const v8f& hi, int col) {
      const int c = col >> 4, uu = col & 15;
      const float bb = um3b[col];
#pragma unroll
      for (int r = 0; r < 8; ++r) {
        int row = rb + r;
        out[(((size_t)(b * Nn + row) * Tt + t) * C + c) * F + G + R + uu] =
            tanhf(lo[r] + bb);
        row += 16;
        if (row < Nn)
          out[(((size_t)(b * Nn + row) * Tt + t) * C + c) * F + G + R + uu] =
              tanhf(hi[r] + bb);
      }
    };
    write_u(c00, c10, p * 32 + n16);
    write_u(c01, c11, p * 32 + 16 + n16);
  }
}

// -------- relation paths: one block per (bt, receiver i); 24 senders --------
__global__ __launch_bounds__(256) void koop_rel_kernel(
    const float* __restrict__ states,
    const float* __restrict__ rs1w, const float* __restrict__ rs1b,
    const float* __restrict__ rs2w, const float* __restrict__ rs2b,
    const float* __restrict__ rs3w, const float* __restrict__ rs3b,
    const float* __restrict__ rm1w, const float* __restrict__ rm1b,
    const float* __restrict__ rm2w, const float* __restrict__ rm2b,
    const float* __restrict__ rm3w, const float* __restrict__ rm3b,
    float* __restrict__ out, float* __restrict__ sel_out) {
  __shared__ __align__(16) float rx[32 * 48];
  __shared__ __align__(16) float hA[32 * 512];
  __shared__ __align__(16) float hB[32 * 512];
  __shared__ __align__(16) float red[256];

  const int tid = threadIdx.x;
  const int wave = tid >> 5;
  const int lane = tid & 31;
  const int n16 = lane & 15;
  const int rb = (lane >> 4) * 8;
  const int blk = blockIdx.x;
  const int bt = blk / Nn;
  const int i = blk - bt * Nn;  // receiver
  const int b = bt >> 4, t = bt & 15;

  for (int idx = tid; idx < 32 * 48; idx += 256) {   // rel_x rows: [x_i | x_j | x_j-x_i]
    int j = idx / 48, col = idx - j * 48;
    float v = 0.f;
    if (j < Nn) {
      int sd = col & 15;
      float xi = states[((size_t)(b * Nn + i) * Tt + t) * SD + sd];
      float xj = states[((size_t)(b * Nn + j) * Tt + t) * SD + sd];
      v = (col < 16) ? xi : (col < 32) ? xj : (xj - xi);
    }
    rx[idx] = v;
  }
  __syncthreads();
  for (int p = wave; p < 8; p += 8) {                // rsel1: 48 -> 256
    v8f c00 = {}, c01 = {}, c10 = {}, c11 = {};
    wave_gemm22<48>(rx, 48, rs1w, 256, p * 32, c00, c01, c10, c11);
    epi_relu_lds(hA, 256, p * 32, rs1b, c00, c10);
    epi_relu_lds(hA, 256, p * 32 + 16, rs1b, c01, c11);
  }
  __syncthreads();
  for (int p = wave; p < 8; p += 8) {                // rsel2: 256 -> 256
    v8f c00 = {}, c01 = {}, c10 = {}, c11 = {};
    wave_gemm22<256>(hA, 256, rs2w, 256, p * 32, c00, c01, c10, c11);
    epi_relu_lds(hB, 256, p * 32, rs2b, c00, c10);
    epi_relu_lds(hB, 256, p * 32 + 16, rs2b, c01, c11);
  }
  __syncthreads();
  {                                                   // rsel3 dot -> hard sigmoid
    int row = tid >> 3, seg = tid & 7;
    const float* hr = hB + row * 256 + seg * 32;
    const float* wr = rs3w + seg * 32;
    float s = 0.f;
#pragma unroll
    for (int q = 0; q < 32; ++q) s += hr[q] * wr[q];
    red[tid] = s;
  }
  __syncthreads();
  if (tid < Nn) {
    float tot = rs3b[0];
#pragma unroll
    for (int q = 0; q < 8; ++q) tot += red[tid * 8 + q];
    float sig = 1.f / (1.f + expf(-tot));
    sel_out[((size_t)(b * Nn + i) * Tt + t) * Nn + tid] = rintf(sig);
  }
  __syncthreads();
  for (int p = wave; p < 16; p += 8) {               // rm1: 48 -> 512
    v8f c00 = {}, c01 = {}, c10 = {}, c11 = {};
    wave_gemm22<48>(rx, 48, rm1w, 512, p * 32, c00, c01, c10, c11);
    epi_relu_lds(hA, 512, p * 32, rm1b, c00, c10);
    epi_relu_lds(hA, 512, p * 32 + 16, rm1b, c01, c11);
  }
  __syncthreads();
  for (int p = wave; p < 16; p += 8) {               // rm2: 512 -> 512
    v8f c00 = {}, c01 = {}, c10 = {}, c11 = {};
    wave_gemm22<512>(hA, 512, rm2w, 512, p * 32, c00, c01, c10, c11);
    epi_relu_lds(hB, 512, p * 32, rm2b, c00, c10);
    epi_relu_lds(hB, 512, p * 32 + 16, rm2b, c01, c11);
  }
  __syncthreads();
  for (int p = wave; p < 8; p += 8) {                // rm3: 512 -> 256 (rel_enc)
    v8f c00 = {}, c01 = {}, c10 = {}, c11 = {};
    wave_gemm22<512>(hB, 512, rm3w, 256, p * 32, c00, c01, c10, c11);
    auto store_enc = [&](const v8f& lo, const v8f& hi, int col) {
      const float bb = rm3b[col];
#pragma unroll
      for (int r = 0; r < 8; ++r) {
        hA[(rb + r) * 256 + col] = lo[r] + bb;
        hA[(16 + rb + r) * 256 + col] = hi[r] + bb;
      }
    };
    store_enc(c00, c10, p * 32 + n16);
    store_enc(c01, c11, p * 32 + 16 + n16);
  }
  __syncthreads();
  {                                                   // masked sum over senders
    const int col = tid;
    float s = 0.f;
#pragma unroll
    for (int j = 0; j < Nn; ++j)
      if (j != i) s += hA[j * 256 + col];
    const int c = col >> 6, r = col & 63;
    out[(((size_t)(b * Nn + i) * Tt + t) * C + c) * F + G + r] = s;
  }
}

extern "C" void kernel_launch(void* const* d_in, const int* in_sizes, int n_in,
                              void* d_out, int out_size, void* d_ws, size_t ws_size,
                              hipStream_t stream) {
  (void)in_sizes; (void)n_in; (void)out_size; (void)d_ws; (void)ws_size;
  const float* states = (const float*)d_in[0];
  const float* e1w = (const float*)d_in[1];  const float* e1b = (const float*)d_in[2];
  const float* e2w = (const float*)d_in[3];  const float* e2b = (const float*)d_in[4];
  const float* e3w = (const float*)d_in[5];  const float* e3b = (const float*)d_in[6];
  const float* e4w = (const float*)d_in[7];  const float* e4b = (const float*)d_in[8];
  const float* us1w = (const float*)d_in[9];  const float* us1b = (const float*)d_in[10];
  const float* us2w = (const float*)d_in[11]; const float* us2b = (const float*)d_in[12];
  const float* um1w = (const float*)d_in[13]; const float* um1b = (const float*)d_in[14];
  const float* um2w = (const float*)d_in[15]; const float* um2b = (const float*)d_in[16];
  const float* um3w = (const float*)d_in[17]; const float* um3b = (const float*)d_in[18];
  const float* rs1w = (const float*)d_in[19]; const float* rs1b = (const float*)d_in[20];
  const float* rs2w = (const float*)d_in[21]; const float* rs2b = (const float*)d_in[22];
  const float* rs3w = (const float*)d_in[23]; const float* rs3b = (const float*)d_in[24];
  const float* rm1w = (const float*)d_in[25]; const float* rm1b = (const float*)d_in[26];
  const float* rm2w = (const float*)d_in[27]; const float* rm2b = (const float*)d_in[28];
  const float* rm3w = (const float*)d_in[29]; const float* rm3b = (const float*)d_in[30];

  float* out = (float*)d_out;
  float* sel = out + OUT_ELEMS;
  float* ug = sel + SEL_ELEMS;

  koop_node_kernel<<<dim3(Bb * Tt), dim3(256), 0, stream>>>(
      states, e1w, e1b, e2w, e2b, e3w, e3b, e4w, e4b, us1w, us1b, us2w, us2b,
      um1w, um1b, um2w, um2b, um3w, um3b, out, ug);
  koop_rel_kernel<<<dim3(Bb * Tt * Nn), dim3(256), 0, stream>>>(
      states, rs1w, rs1b, rs2w, rs2b, rs3w, rs3b, rm1w, rm1b, rm2w, rm2b,
      rm3w, rm3b, out, sel);
}